// OnlineHadamard_39934605918508
// MI455X (gfx1250) — compile-verified
//
#include <hip/hip_runtime.h>
#include <hip/hip_bf16.h>

typedef __attribute__((ext_vector_type(2))) float v2f;
typedef __attribute__((ext_vector_type(4))) float v4f;
typedef __attribute__((ext_vector_type(8))) float v8f;

#define J_STRIDE 20        // padded inner-row stride (dwords): m*20 mod 64 all-distinct -> conflict-free
#define I_STRIDE 336       // padded slice stride = 16*20 + 16 (keeps half-wave banks disjoint in stage 3)

__device__ __forceinline__ float hsign(int a, int b) {
    return (__popc(a & b) & 1) ? -1.0f : 1.0f;
}

// One block = one row of 4096 f32. Row viewed as X[i][j][k], i,j,k in [0,16).
// FWHT_4096 = H16 applied along k, then j, then i (H_4096 = H16 (x) H16 (x) H16).
// Each 16x16 slice transform = 4 chained V_WMMA_F32_16X16X4_F32 accumulations.
__global__ __launch_bounds__(256)
void fwht4096_wmma_kernel(const float* __restrict__ x, float* __restrict__ out) {
    __shared__ __align__(16) float lds[16 * I_STRIDE];   // 5376 floats = 21 KB

    const float* xr = x + (size_t)blockIdx.x * 4096;
    float* orow     = out + (size_t)blockIdx.x * 4096;

    const int t = threadIdx.x;

    // Load row, fully coalesced: each pass, 256 threads fetch 256 contiguous
    // float4 (4 KB). Padded-LDS targets are 16B aligned -> ds_store_b128.
    #pragma unroll
    for (int p = 0; p < 4; ++p) {
        const int e = p * 1024 + t * 4;                 // flat element index
        const int i = e >> 8, j = (e >> 4) & 15, k = e & 15;
        const v4f v = __builtin_nontemporal_load((const v4f*)(xr + e));
        *(v4f*)&lds[i * I_STRIDE + j * J_STRIDE + k] = v;
    }
    __syncthreads();

    const int lane = t & 31;
    const int wave = t >> 5;
    const int half = lane >> 4;   // which half-wave
    const int lm   = lane & 15;

    // H16 fragment: element (lm, k) with k = c*4 + 2*half + v.
    // H is symmetric, so the same registers serve as A-layout H[m][k]
    // (m=lm) and B-layout H[k][n] (n=lm).
    v2f hfrag[4];
    #pragma unroll
    for (int c = 0; c < 4; ++c) {
        const int k0 = c * 4 + 2 * half;
        hfrag[c][0] = hsign(lm, k0);
        hfrag[c][1] = hsign(lm, k0 + 1);
    }

    // ---- Stage 1: per slice i: T1[i] = X[i] * H   (transform along k) ----
    #pragma unroll
    for (int s = 0; s < 2; ++s) {
        float* base = &lds[(wave * 2 + s) * I_STRIDE];
        v2f a[4];
        #pragma unroll
        for (int c = 0; c < 4; ++c) {
            const int k0 = c * 4 + 2 * half;
            a[c][0] = base[lm * J_STRIDE + k0];
            a[c][1] = base[lm * J_STRIDE + k0 + 1];
        }
        v8f d = {};
        #pragma unroll
        for (int c = 0; c < 4; ++c)
            d = __builtin_amdgcn_wmma_f32_16x16x4_f32(false, a[c], false, hfrag[c],
                                                      (short)0, d, false, false);
        #pragma unroll
        for (int v = 0; v < 8; ++v)
            base[(8 * half + v) * J_STRIDE + lm] = d[v];
    }

    // Slice is wave-private and LDS ops are in-order per wave: no barrier needed.
    // ---- Stage 2: per slice i: T2[i] = H * T1[i]  (transform along j) ----
    #pragma unroll
    for (int s = 0; s < 2; ++s) {
        float* base = &lds[(wave * 2 + s) * I_STRIDE];
        v2f b[4];
        #pragma unroll
        for (int c = 0; c < 4; ++c) {
            const int k0 = c * 4 + 2 * half;
            b[c][0] = base[k0 * J_STRIDE + lm];
            b[c][1] = base[(k0 + 1) * J_STRIDE + lm];
        }
        v8f d = {};
        #pragma unroll
        for (int c = 0; c < 4; ++c)
            d = __builtin_amdgcn_wmma_f32_16x16x4_f32(false, hfrag[c], false, b[c],
                                                      (short)0, d, false, false);
        #pragma unroll
        for (int v = 0; v < 8; ++v)
            base[(8 * half + v) * J_STRIDE + lm] = d[v];
    }
    __syncthreads();   // stage 3 reads across all slices

    // ---- Stage 3: per slice j: out[:,j,:] = (1/64) * H * T2[:,j,:]  (transform along i) ----
    #pragma unroll
    for (int s = 0; s < 2; ++s) {
        const int j = wave * 2 + s;
        v2f b[4];
        #pragma unroll
        for (int c = 0; c < 4; ++c) {
            const int k0 = c * 4 + 2 * half;
            b[c][0] = lds[k0 * I_STRIDE + j * J_STRIDE + lm];
            b[c][1] = lds[(k0 + 1) * I_STRIDE + j * J_STRIDE + lm];
        }
        v8f d = {};
        #pragma unroll
        for (int c = 0; c < 4; ++c)
            d = __builtin_amdgcn_wmma_f32_16x16x4_f32(false, hfrag[c], false, b[c],
                                                      (short)0, d, false, false);
        #pragma unroll
        for (int v = 0; v < 8; ++v)
            __builtin_nontemporal_store(d[v] * 0.015625f,   // 1/64 = 1/sqrt(4096)
                                        &orow[(8 * half + v) * 256 + j * 16 + lm]);
    }
}

extern "C" void kernel_launch(void* const* d_in, const int* in_sizes, int n_in,
                              void* d_out, int out_size, void* d_ws, size_t ws_size,
                              hipStream_t stream) {
    const float* x = (const float*)d_in[0];
    float* out = (float*)d_out;
    const int rows = in_sizes[0] / 4096;   // 2 * 4096 = 8192
    if (rows > 0) {
        fwht4096_wmma_kernel<<<rows, 256, 0, stream>>>(x, out);
    }
}